// SparseMLP_83846351553053
// MI455X (gfx1250) — compile-verified
//
#include <hip/hip_runtime.h>
#include <stdint.h>

#define DM   512
#define DFF  2048
#define NE   8
#define CAP  10240
#define TTOK 65536
#define MT   32      // capacity rows per FFN workgroup
#define FC   64      // F-chunk

typedef __attribute__((ext_vector_type(2))) float v2f;
typedef __attribute__((ext_vector_type(8))) float v8f;

// ---------------------------------------------------------------------------
// Kernel 1: router. One block = 256 tokens (8 waves, wave-per-token loop x32).
// Writes probs [T,8], top1 [T], expert_id [T]; per-block pi partials & counts
// (deterministic: per-wave LDS accumulation, fixed-order combine).
// ---------------------------------------------------------------------------
__global__ __launch_bounds__(256) void k_router(
    const float* __restrict__ x, const float* __restrict__ Wr,
    const float* __restrict__ br, float* __restrict__ probs,
    float* __restrict__ top1, int* __restrict__ eid,
    float* __restrict__ blockPi, int* __restrict__ blockCnt)
{
    __shared__ float wr[NE * DM];
    __shared__ float bsh[NE];
    __shared__ float pi8[8][NE];
    __shared__ int   cnt8[8][NE];

    int tid = threadIdx.x;
    for (int i = tid; i < NE * DM; i += 256) wr[i] = Wr[i];
    if (tid < NE) bsh[tid] = br[tid];
    if (tid < 64) { ((float*)pi8)[tid] = 0.f; ((int*)cnt8)[tid] = 0; }
    __syncthreads();

    int wave = tid >> 5, lane = tid & 31;
    int base = blockIdx.x * 256 + wave * 32;

    for (int i = 0; i < 32; ++i) {
        int t = base + i;
        const float4* xr = (const float4*)(x + (size_t)t * DM) + lane * 4;
        float4 xv0 = xr[0], xv1 = xr[1], xv2 = xr[2], xv3 = xr[3];

        float lg[NE];
        #pragma unroll
        for (int e = 0; e < NE; ++e) {
            const float4* wrow = (const float4*)(wr + e * DM) + lane * 4;
            float4 w0 = wrow[0], w1 = wrow[1], w2 = wrow[2], w3 = wrow[3];
            float s = xv0.x*w0.x + xv0.y*w0.y + xv0.z*w0.z + xv0.w*w0.w
                    + xv1.x*w1.x + xv1.y*w1.y + xv1.z*w1.z + xv1.w*w1.w
                    + xv2.x*w2.x + xv2.y*w2.y + xv2.z*w2.z + xv2.w*w2.w
                    + xv3.x*w3.x + xv3.y*w3.y + xv3.z*w3.z + xv3.w*w3.w;
            #pragma unroll
            for (int off = 16; off >= 1; off >>= 1) s += __shfl_xor(s, off, 32);
            lg[e] = s + bsh[e];
        }
        // softmax + argmax (first-max tie-break, matching jnp.argmax)
        float m = lg[0]; int bi = 0;
        #pragma unroll
        for (int e = 1; e < NE; ++e) if (lg[e] > m) { m = lg[e]; bi = e; }
        float den = 0.f, p[NE];
        #pragma unroll
        for (int e = 0; e < NE; ++e) { p[e] = __expf(lg[e] - m); den += p[e]; }
        float inv = 1.f / den;
        float pb = 0.f;
        #pragma unroll
        for (int e = 0; e < NE; ++e) { p[e] *= inv; if (e == bi) pb = p[e]; }

        if (lane < NE) {
            probs[(size_t)t * NE + lane] = p[lane];
            pi8[wave][lane] += p[lane];          // wave-sequential, deterministic
        }
        if (lane == 0) {
            top1[t] = pb;
            eid[t]  = bi;
            cnt8[wave][bi] += 1;
        }
    }
    __syncthreads();
    if (tid < NE) {
        float s = 0.f; int c = 0;
        #pragma unroll
        for (int w = 0; w < 8; ++w) { s += pi8[w][tid]; c += cnt8[w][tid]; }
        blockPi[blockIdx.x * NE + tid]  = s;
        blockCnt[blockIdx.x * NE + tid] = c;
    }
}

// ---------------------------------------------------------------------------
// Kernel 2: sequential inter-block scan (256 blocks x 8 experts), totals,
// kept counts, aux loss, num_dropped. Tiny, fixed-order => deterministic.
// ---------------------------------------------------------------------------
__global__ __launch_bounds__(64) void k_scan(
    const int* __restrict__ blockCnt, const float* __restrict__ blockPi,
    int* __restrict__ blockOff, int* __restrict__ cntE,
    float* __restrict__ aux_out, float* __restrict__ nd_out)
{
    __shared__ int   tot[NE];
    __shared__ float pis[NE];
    int tid = threadIdx.x;
    if (tid < NE) {
        int run = 0; float ps = 0.f;
        for (int b = 0; b < 256; ++b) {
            blockOff[b * NE + tid] = run;
            run += blockCnt[b * NE + tid];
            ps  += blockPi[b * NE + tid];
        }
        tot[tid] = run;
        pis[tid] = ps;
        cntE[tid] = run < CAP ? run : CAP;
    }
    __syncthreads();
    if (tid == 0) {
        float aux = 0.f; int nd = 0;
        for (int e = 0; e < NE; ++e) {
            int ke = tot[e] < CAP ? tot[e] : CAP;
            nd += tot[e] - ke;
            aux += ((float)ke / (float)TTOK) * (pis[e] / (float)TTOK);
        }
        *aux_out = (float)NE * aux;
        *nd_out  = (float)nd;
    }
}

// ---------------------------------------------------------------------------
// Kernel 3: dispatch. In-block inclusive scan of packed 8x u16 expert one-hots
// (token order preserved -> matches jnp.cumsum priority). Builds slot->token,
// writes one-hot rows, zeroes output rows of dropped tokens.
// ---------------------------------------------------------------------------
__global__ __launch_bounds__(256) void k_dispatch(
    const int* __restrict__ eid, const int* __restrict__ blockOff,
    int* __restrict__ slot2tok, float* __restrict__ onehot,
    float* __restrict__ out)
{
    __shared__ uint32_t sc[4][256];
    int tid = threadIdx.x;
    int t   = blockIdx.x * 256 + tid;
    int e   = eid[t];

    uint32_t my[4] = {0u, 0u, 0u, 0u};
    my[e >> 1] = 1u << (16 * (e & 1));
    sc[0][tid] = my[0]; sc[1][tid] = my[1]; sc[2][tid] = my[2]; sc[3][tid] = my[3];
    __syncthreads();
    for (int d = 1; d < 256; d <<= 1) {
        uint32_t v0 = 0, v1 = 0, v2 = 0, v3 = 0;
        if (tid >= d) { v0 = sc[0][tid-d]; v1 = sc[1][tid-d]; v2 = sc[2][tid-d]; v3 = sc[3][tid-d]; }
        __syncthreads();
        sc[0][tid] += v0; sc[1][tid] += v1; sc[2][tid] += v2; sc[3][tid] += v3;
        __syncthreads();
    }
    uint32_t word = sc[e >> 1][tid];
    int rank = (e & 1) ? (int)(word >> 16) : (int)(word & 0xffffu);  // inclusive
    int pos  = blockOff[blockIdx.x * NE + e] + rank - 1;
    bool kept = pos < CAP;

    float4* ohp = (float4*)(onehot + (size_t)t * NE);
    float oh[NE];
    #pragma unroll
    for (int j = 0; j < NE; ++j) oh[j] = (kept && j == e) ? 1.f : 0.f;
    ohp[0] = make_float4(oh[0], oh[1], oh[2], oh[3]);
    ohp[1] = make_float4(oh[4], oh[5], oh[6], oh[7]);

    if (kept) {
        slot2tok[e * CAP + pos] = t;
    } else {
        float4 z = make_float4(0.f, 0.f, 0.f, 0.f);
        float4* op = (float4*)(out + (size_t)t * DM);
        for (int q = 0; q < DM / 4; ++q) op[q] = z;
    }
}

// ---------------------------------------------------------------------------
// Kernel 4: fused per-expert FFN + combine.
// Block = 8 waves, M=32 slot rows, full D=512 output, F chunked by 64.
//   GEMM1 (per chunk): wave (mt, quad) computes 16x16 h-tile over K=512,
//     A from LDS x-tile, B streamed from Wi (L2-resident). ReLU -> LDS.
//   GEMM2: wave accumulates 8 y-tiles (16x16) over chunk K=64 (8-way ILP).
// Epilogue scales by top-1 prob and scatters rows back to out[token].
// ---------------------------------------------------------------------------
__global__ __launch_bounds__(256) void k_ffn(
    const float* __restrict__ x, const float* __restrict__ Wi,
    const float* __restrict__ Wo, const int* __restrict__ slot2tok,
    const int* __restrict__ cntE, const float* __restrict__ top1,
    float* __restrict__ out)
{
    int e  = blockIdx.y;
    int m0 = blockIdx.x * MT;
    int cnt = cntE[e];
    if (m0 >= cnt) return;                    // uniform early-exit (empty tile)

    __shared__ float xs[MT][DM + 4];          // stride 516: conflict-free A reads
    __shared__ float hs[MT][FC + 4];          // stride 68:  conflict-free A reads
    __shared__ int   tokS[MT];
    __shared__ float pS[MT];

    int tid  = threadIdx.x;
    int wave = tid >> 5, lane = tid & 31;

    if (tid < MT) {
        int r   = m0 + tid;
        int tok = (r < cnt) ? slot2tok[e * CAP + r] : -1;
        tokS[tid] = tok;
        pS[tid]   = (tok >= 0) ? top1[tok] : 0.f;
    }
    __syncthreads();

    // stage X tile: lane = row, wave = 64-column slab
    {
        int r   = lane;
        int c0  = wave * 64;
        int tok = tokS[r];
        if (tok >= 0) {
            const float4* src = (const float4*)(x + (size_t)tok * DM + c0);
            #pragma unroll 4
            for (int q = 0; q < 16; ++q) {
                float4 v = src[q];
                float* dst = &xs[r][c0 + q * 4];
                dst[0] = v.x; dst[1] = v.y; dst[2] = v.z; dst[3] = v.w;
            }
        } else {
            for (int q = 0; q < 64; ++q) xs[r][c0 + q] = 0.f;
        }
    }
    __syncthreads();

    int mt   = wave & 1;                 // M-tile (0..1)
    int quad = wave >> 1;                // 0..3: F-tile in GEMM1, N-group in GEMM2
    int rlo  = lane & 15;                // row/col-within-tile
    int khi  = (lane >> 4) * 2;          // per-lane K offset (layout: i + 2*(lane/16))
    int rhi  = (lane >> 4) * 8;          // C/D layout row offset

    const float* WiE = Wi + (size_t)e * DFF * DM;
    const float* WoE = Wo + (size_t)e * DM * DFF;

    v8f acc[8];
    #pragma unroll
    for (int n = 0; n < 8; ++n) acc[n] = (v8f){0.f,0.f,0.f,0.f,0.f,0.f,0.f,0.f};

    for (int fc = 0; fc < DFF; fc += FC) {
        // ---- GEMM1: one 16x16 h-tile per wave, K = 512 ----
        const float* wiP = WiE + (size_t)(fc + quad * 16 + rlo) * DM + khi;
        const float* xaP = &xs[mt * 16 + rlo][khi];
        v8f h0 = (v8f){0.f,0.f,0.f,0.f,0.f,0.f,0.f,0.f};
        v8f h1 = (v8f){0.f,0.f,0.f,0.f,0.f,0.f,0.f,0.f};
        #pragma unroll 4
        for (int k = 0; k < DM; k += 8) {   // two partial accumulators for ILP
            v2f a0 = *(const v2f*)(xaP + k);
            v2f b0 = *(const v2f*)(wiP + k);
            v2f a1 = *(const v2f*)(xaP + k + 4);
            v2f b1 = *(const v2f*)(wiP + k + 4);
            h0 = __builtin_amdgcn_wmma_f32_16x16x4_f32(false, a0, false, b0,
                                                       (short)0, h0, false, false);
            h1 = __builtin_amdgcn_wmma_f32_16x16x4_f32(false, a1, false, b1,
                                                       (short)0, h1, false, false);
        }
        // ReLU + stash h-tile in LDS (C/D layout -> row-major hs)
        #pragma unroll
        for (int j = 0; j < 8; ++j) {
            float v0 = h0[j] + h1[j];
            hs[mt * 16 + j + rhi][quad * 16 + rlo] = v0 > 0.f ? v0 : 0.f;
        }
        __syncthreads();

        // ---- GEMM2: accumulate 8 y-tiles over chunk K = 64 ----
        const float* haP = &hs[mt * 16 + rlo][khi];
        const float* woP = WoE + (size_t)(quad * 128 + rlo) * DFF + fc + khi;
        #pragma unroll 2
        for (int k = 0; k < FC; k += 4) {
            v2f a = *(const v2f*)(haP + k);
            #pragma unroll
            for (int n = 0; n < 8; ++n) {   // 8 independent WMMAs per k-step
                v2f b = *(const v2f*)(woP + (size_t)n * 16 * DFF + k);
                acc[n] = __builtin_amdgcn_wmma_f32_16x16x4_f32(false, a, false, b,
                                                               (short)0, acc[n], false, false);
            }
        }
        __syncthreads();
    }

    // ---- combine: scale by top-1 prob, scatter to out[token] ----
    #pragma unroll
    for (int n = 0; n < 8; ++n) {
        int col = (quad * 8 + n) * 16 + rlo;
        #pragma unroll
        for (int j = 0; j < 8; ++j) {
            int row = mt * 16 + j + rhi;
            int tok = tokS[row];
            if (tok >= 0) out[(size_t)tok * DM + col] = acc[n][j] * pS[row];
        }
    }
}

// ---------------------------------------------------------------------------
extern "C" void kernel_launch(void* const* d_in, const int* in_sizes, int n_in,
                              void* d_out, int out_size, void* d_ws, size_t ws_size,
                              hipStream_t stream)
{
    const float* x  = (const float*)d_in[0];   // [T, 512]
    const float* Wr = (const float*)d_in[1];   // [8, 512]
    const float* br = (const float*)d_in[2];   // [8]
    const float* Wi = (const float*)d_in[3];   // [8, 2048, 512]
    const float* Wo = (const float*)d_in[4];   // [8, 512, 2048]

    float* outp = (float*)d_out;               // [T, 512]
    float* oh   = outp + 33554432ull;          // onehot     [T, 8]
    float* t1   = oh   + 524288ull;            // top1_probs [T]
    float* pr   = t1   + 65536ull;             // probs      [T, 8]
    float* aux  = pr   + 524288ull;            // aux_loss   scalar
    float* nd   = aux  + 1;                    // num_dropped scalar

    char*  w        = (char*)d_ws;
    int*   eid      = (int*)  (w + 0);         // T ints
    float* blockPi  = (float*)(w + 262144);    // 256 x 8
    int*   blockCnt = (int*)  (w + 270336);    // 256 x 8
    int*   blockOff = (int*)  (w + 278528);    // 256 x 8
    int*   cntE     = (int*)  (w + 286720);    // 8
    int*   slot2tok = (int*)  (w + 287744);    // 8 x 10240

    k_router  <<<256, 256, 0, stream>>>(x, Wr, br, pr, t1, eid, blockPi, blockCnt);
    k_scan    <<<1,    64, 0, stream>>>(blockCnt, blockPi, blockOff, cntE, aux, nd);
    k_dispatch<<<256, 256, 0, stream>>>(eid, blockOff, slot2tok, oh, outp);
    k_ffn     <<<dim3((CAP + MT - 1) / MT, NE), 256, 0, stream>>>(
                  x, Wi, Wo, slot2tok, cntE, t1, outp);
}